// UNetGCN_20151986553226
// MI455X (gfx1250) — compile-verified
//
#include <hip/hip_runtime.h>

// ---------------------------------------------------------------------------
// UNet-GCN on MI455X (gfx1250), fp32 end-to-end.
// GEMMs: V_WMMA_F32_16X16X4_F32, 4 waves/block share a 16xK A-panel staged in
// LDS (coalesced b128 global loads -> conflict-free ds_load_b64 fragments),
// with global_prefetch of the next panel. Edge aggregation: wave-per-edge
// float4 gather + global f32 atomics (L2-resident; 192MB L2 holds the model).
// ---------------------------------------------------------------------------

typedef float v2f __attribute__((ext_vector_type(2)));
typedef float v8f __attribute__((ext_vector_type(8)));

#define FLAG_BIAS 1
#define FLAG_ADD  2
#define FLAG_GELU 4

#define KC 32          // K-chunk staged in LDS
#define NW 4           // waves per block (16x64 output tile)

__device__ __forceinline__ float gelu_tanh(float x) {
  float x3 = x * x * x;
  return 0.5f * x * (1.0f + tanhf(0.7978845608028654f * (x + 0.044715f * x3)));
}

// Generic strided GEMM: C[M,Nc] = A[M,K] * B + epilogue.
//   A: row-major, leading dim lda.
//   B: element (k, n) at B[k*sK + n*sN]  (handles transposed / interleaved W).
//   C: row-major, leading dim ldc (write into column sub-slices).
// Block = 4 waves = 16 rows x 64 cols. A 16xKC panel is staged in LDS once
// per block and shared by all 4 waves. Fragment layout per CDNA5 ISA 7.12.2:
//   A 16x4 : lane = half*16 + m ; v2f = A[m][half*2 + {0,1}]
//   B 4x16 : lane = half*16 + n ; v2f = B[half*2 + {0,1}][n]
//   C 16x16: lane col n = lane&15 ; vgpr v -> row m = v + 8*(lane>>4)
__global__ __launch_bounds__(32 * NW)
void gemm_f32_wmma(const float* __restrict__ A, int lda,
                   const float* __restrict__ B, int sK, int sN,
                   float* __restrict__ C, int ldc,
                   int M, int Nc, int K,
                   const float* __restrict__ bias,
                   const float* __restrict__ addm, int ldadd,
                   int flags)
{
  __shared__ float As[16][KC + 4];   // +4 pad -> conflict-free ds_load_b64

  const int tm   = blockIdx.x * 16;
  const int tn0  = blockIdx.y * (16 * NW);
  const int tid  = threadIdx.x;
  const int wave = tid >> 5;
  const int lane = tid & 31;
  const int half = lane >> 4;   // 0 or 1
  const int lm   = lane & 15;
  const int col  = tn0 + wave * 16 + lm;

  // cooperative A-panel loader coords: 128 threads x float4 = 16x32 floats
  const int lr = tid >> 3;          // row 0..15
  const int lc = (tid & 7) * 4;     // col group 0,4,...,28

  v8f acc = {0.f, 0.f, 0.f, 0.f, 0.f, 0.f, 0.f, 0.f};

  for (int kc = 0; kc < K; kc += KC) {
    // stage A[tm..tm+15][kc..kc+KC) into LDS (coalesced 16B loads)
    {
      const float4 v = *(const float4*)(A + (size_t)(tm + lr) * lda + kc + lc);
      As[lr][lc + 0] = v.x;
      As[lr][lc + 1] = v.y;
      As[lr][lc + 2] = v.z;
      As[lr][lc + 3] = v.w;
    }
    // prefetch next panel while this chunk computes
    if (kc + KC < K)
      __builtin_prefetch(A + (size_t)(tm + lr) * lda + kc + KC + lc, 0, 3);
    __syncthreads();

#pragma unroll
    for (int k = 0; k < KC; k += 4) {
      const int kb = k + half * 2;
      const int kg = kc + kb;
      v2f a, b;
      a.x = As[lm][kb];
      a.y = As[lm][kb + 1];
      b.x = B[(size_t)kg * sK + (size_t)col * sN];
      b.y = B[(size_t)(kg + 1) * sK + (size_t)col * sN];
      acc = __builtin_amdgcn_wmma_f32_16x16x4_f32(
          /*neg_a=*/false, a, /*neg_b=*/false, b,
          /*c_mod=*/(short)0, acc, /*reuse_a=*/false, /*reuse_b=*/false);
    }
    __syncthreads();
  }

  const float bval = (flags & FLAG_BIAS) ? bias[col] : 0.0f;
#pragma unroll
  for (int v = 0; v < 8; ++v) {
    const int row = tm + half * 8 + v;
    if (row < M) {
      float val = acc[v] + bval;
      if (flags & FLAG_ADD)  val += addm[(size_t)row * ldadd + col];
      if (flags & FLAG_GELU) val = gelu_tanh(val);
      C[(size_t)row * ldc + col] = val;
    }
  }
}

// ---------------- degree / norm ----------------
__global__ __launch_bounds__(256)
void deg_init(float* __restrict__ deg, int n) {
  int i = blockIdx.x * blockDim.x + threadIdx.x;
  if (i < n) deg[i] = 1.0f;  // self-loop
}

__global__ __launch_bounds__(256)
void deg_count(const int* __restrict__ tgt, float* __restrict__ deg, int e) {
  int i = blockIdx.x * blockDim.x + threadIdx.x;
  if (i < e) atomicAdd(&deg[tgt[i]], 1.0f);
}

__global__ __launch_bounds__(256)
void deg_rsqrt(float* __restrict__ deg, int n) {
  int i = blockIdx.x * blockDim.x + threadIdx.x;
  if (i < n) deg[i] = rsqrtf(deg[i]);
}

// ---------------- GCN aggregation ----------------
// Self-loop term: agg[n][d] = hw[n][d] * dinv[n]^2
__global__ __launch_bounds__(256)
void agg_init(const float* __restrict__ hw, const float* __restrict__ dinv,
              float* __restrict__ agg, int n, int d) {
  int i = blockIdx.x * blockDim.x + threadIdx.x;
  if (i < (int)((long long)n * d)) {
    float dv = dinv[i / d];
    agg[i] = hw[i] * dv * dv;
  }
}

// One wave per edge: lanes cover the feature dim with float4 gathers and
// per-float atomic adds into the target row (L2-resident traffic).
__global__ __launch_bounds__(256)
void edge_scatter(const float* __restrict__ hw, const int* __restrict__ src,
                  const int* __restrict__ tgt, const float* __restrict__ dinv,
                  float* __restrict__ agg, int e, int d) {
  int wid  = (blockIdx.x * blockDim.x + threadIdx.x) >> 5;
  int lane = threadIdx.x & 31;
  if (wid >= e) return;
  const int s = src[wid];
  const int t = tgt[wid];
  const float nrm = dinv[s] * dinv[t];
  const float* __restrict__ hrow = hw  + (size_t)s * d;
  float* __restrict__       arow = agg + (size_t)t * d;
  for (int c = lane * 4; c < d; c += 128) {
    float4 v = *(const float4*)(hrow + c);
    atomicAdd(arow + c + 0, v.x * nrm);
    atomicAdd(arow + c + 1, v.y * nrm);
    atomicAdd(arow + c + 2, v.z * nrm);
    atomicAdd(arow + c + 3, v.w * nrm);
  }
}

// out = gelu(agg + bias)
__global__ __launch_bounds__(256)
void bias_gelu(const float* __restrict__ agg, const float* __restrict__ bias,
               float* __restrict__ out, int n, int d) {
  int i = blockIdx.x * blockDim.x + threadIdx.x;
  if (i < (int)((long long)n * d)) out[i] = gelu_tanh(agg[i] + bias[i % d]);
}

// Rearrange fin_w [64, 512] (col j = c*8 + p) into Bbig [512, 64] with
// row r = p*64 + c, matching the slice-major d3 layout.
__global__ __launch_bounds__(256)
void build_fin(const float* __restrict__ fin_w, float* __restrict__ Bbig) {
  int i = blockIdx.x * blockDim.x + threadIdx.x;  // 512*64
  if (i < 512 * 64) {
    int o = i & 63;
    int r = i >> 6;
    int p = r >> 6;     // 0..7
    int c = r & 63;     // 0..63
    Bbig[(size_t)r * 64 + o] = fin_w[(size_t)o * 512 + c * 8 + p];
  }
}

// ---------------------------------------------------------------------------
static inline void launch_gemm(hipStream_t s,
                               const float* A, int lda,
                               const float* B, int sK, int sN,
                               float* C, int ldc, int M, int Nc, int K,
                               const float* bias, const float* addm, int ldadd,
                               int flags) {
  dim3 grid((unsigned)(M / 16), (unsigned)(Nc / (16 * NW)));
  gemm_f32_wmma<<<grid, dim3(32 * NW), 0, s>>>(A, lda, B, sK, sN, C, ldc,
                                               M, Nc, K, bias, addm, ldadd, flags);
}

extern "C" void kernel_launch(void* const* d_in, const int* in_sizes, int n_in,
                              void* d_out, int out_size, void* d_ws, size_t ws_size,
                              hipStream_t stream) {
  const int IN_D = 128, H = 64;
  const int N = in_sizes[0] / IN_D;       // 50000 (multiple of 16)
  const int E = in_sizes[1] / 2;          // 800000

  const float* x        = (const float*)d_in[0];
  const int*   eidx     = (const int*)d_in[1];
  const float* W1       = (const float*)d_in[2];
  const float* b1       = (const float*)d_in[3];
  const float* W2       = (const float*)d_in[4];
  const float* b2       = (const float*)d_in[5];
  const float* W3       = (const float*)d_in[6];
  const float* b3       = (const float*)d_in[7];
  const float* dec1_w   = (const float*)d_in[8];   // [256,128,2]
  const float* dec1_b   = (const float*)d_in[9];
  const float* dec2_w   = (const float*)d_in[10];  // [128,64,2]
  const float* dec2_b   = (const float*)d_in[11];
  const float* dec3_w   = (const float*)d_in[12];  // [64,64,2]
  const float* dec3_b   = (const float*)d_in[13];
  const float* skip1_w  = (const float*)d_in[14];  // [128,128]
  const float* skip1_b  = (const float*)d_in[15];
  const float* skip2_w  = (const float*)d_in[16];  // [64,64]
  const float* skip2_b  = (const float*)d_in[17];
  const float* fin_w    = (const float*)d_in[18];  // [64,512]
  const float* fin_b    = (const float*)d_in[19];
  float* out = (float*)d_out;

  const int* src = eidx;       // edge_index[0]
  const int* tgt = eidx + E;   // edge_index[1]

  // ---- workspace carve-up (floats) ----
  float* w = (float*)d_ws;
  size_t o = 0;
  float* dinv = w + o; o += (size_t)N;          // deg -> dinv in place
  float* tmpA = w + o; o += (size_t)N * 256;    // hw scratch
  float* tmpB = w + o; o += (size_t)N * 256;    // agg / skip scratch
  float* x1   = w + o; o += (size_t)N * 64;
  float* x2   = w + o; o += (size_t)N * 128;
  float* x3d2 = w + o; o += (size_t)N * 256;    // x3, later reused for d2
  float* d1   = w + o; o += (size_t)N * 256;
  float* d3   = w + o; o += (size_t)N * 512;
  float* Bbig = w + o; o += (size_t)512 * 64;
  (void)ws_size; (void)n_in; (void)out_size;

  const int TB = 256;
  auto blocks = [](long long n, int tb) { return (unsigned)((n + tb - 1) / tb); };

  // ---- degree / normalization ----
  deg_init <<<blocks(N, TB), TB, 0, stream>>>(dinv, N);
  deg_count<<<blocks(E, TB), TB, 0, stream>>>(tgt, dinv, E);
  deg_rsqrt<<<blocks(N, TB), TB, 0, stream>>>(dinv, N);

  // ---- GCN layer helper: hw -> agg -> gelu(agg+b) ----
  auto gcn_layer = [&](const float* hin, int Kd, const float* W, const float* b,
                       int Dd, float* hout) {
    launch_gemm(stream, hin, Kd, W, Dd, 1, tmpA, Dd, N, Dd, Kd,
                nullptr, nullptr, 0, 0);
    agg_init<<<blocks((long long)N * Dd, TB), TB, 0, stream>>>(tmpA, dinv, tmpB, N, Dd);
    edge_scatter<<<blocks((long long)E * 32, TB), TB, 0, stream>>>(
        tmpA, src, tgt, dinv, tmpB, E, Dd);
    bias_gelu<<<blocks((long long)N * Dd, TB), TB, 0, stream>>>(tmpB, b, hout, N, Dd);
  };

  gcn_layer(x,  IN_D, W1, b1, H,     x1);    // [N,64]
  gcn_layer(x1, H,    W2, b2, 2 * H, x2);    // [N,128]
  gcn_layer(x2, 2*H,  W3, b3, 4 * H, x3d2);  // [N,256]  (x3)

  // ---- decoder ----
  // skip1 = x2 @ skip1_w.T + skip1_b   (B transposed: sK=1, sN=128)
  launch_gemm(stream, x2, 128, skip1_w, 1, 128, tmpB, 128, N, 128, 128,
              skip1_b, nullptr, 0, FLAG_BIAS);
  // d1 slice k: x3 @ dec1_w[:,:,k] + dec1_b + skip1  -> d1[:, k*128 : ...]
  for (int k = 0; k < 2; ++k)
    launch_gemm(stream, x3d2, 256, dec1_w + k, 256, 2, d1 + k * 128, 256,
                N, 128, 256, dec1_b, tmpB, 128, FLAG_BIAS | FLAG_ADD);

  // skip2 = x1 @ skip2_w.T + skip2_b
  launch_gemm(stream, x1, 64, skip2_w, 1, 64, tmpB, 64, N, 64, 64,
              skip2_b, nullptr, 0, FLAG_BIAS);
  // d2 slice (l,k): d1[:, l*128:] @ dec2_w[:,:,k] + dec2_b + skip2
  for (int l = 0; l < 2; ++l)
    for (int k = 0; k < 2; ++k)
      launch_gemm(stream, d1 + l * 128, 256, dec2_w + k, 128, 2,
                  x3d2 + (2 * l + k) * 64, 256, N, 64, 128,
                  dec2_b, tmpB, 64, FLAG_BIAS | FLAG_ADD);

  // d3 slice (l,k): d2[:, l*64:] @ dec3_w[:,:,k] + dec3_b
  for (int l = 0; l < 4; ++l)
    for (int k = 0; k < 2; ++k)
      launch_gemm(stream, x3d2 + l * 64, 256, dec3_w + k, 128, 2,
                  d3 + (2 * l + k) * 64, 512, N, 64, 64,
                  dec3_b, nullptr, 0, FLAG_BIAS);

  // final: out = d3 @ Bbig + fin_b
  build_fin<<<blocks(512 * 64, TB), TB, 0, stream>>>(fin_w, Bbig);
  launch_gemm(stream, d3, 512, Bbig, 64, 1, out, 64, N, 64, 512,
              fin_b, nullptr, 0, FLAG_BIAS);
}